// SignDenoising2_72146860638718
// MI455X (gfx1250) — compile-verified
//
#include <hip/hip_runtime.h>

// GIN 2-layer GNN for MI455X (gfx1250, wave32).
// Pipeline per layer:
//   1) scale:   hpre = (1+eps)*x                  (float4 streaming)
//   2) scatter: hpre[dst] += x[src]               (global_atomic_add_f32, L2-resident)
//   3) mlp:     out = relu(hpre@Wa+ba)@Wb+bb      (V_WMMA_F32_16X16X4_F32)

typedef __attribute__((ext_vector_type(2))) float v2f;
typedef __attribute__((ext_vector_type(8))) float v8f;

// ---------------------------------------------------------------- scale
__global__ __launch_bounds__(256, 1)
void gin_scale_kernel(const float* __restrict__ in, const float* __restrict__ eps,
                      float* __restrict__ out, int nvec4) {
    const int i = blockIdx.x * 256 + threadIdx.x;
    if (i < nvec4) {
        const float s = 1.0f + eps[0];
        float4 v = ((const float4*)in)[i];
        v.x *= s; v.y *= s; v.z *= s; v.w *= s;
        ((float4*)out)[i] = v;
    }
}

// ---------------------------------------------------------------- scatter-add
// One thread handles (edge, 2 consecutive channels). 32 threads cover one
// edge's 64-float row. Atomics land in L2 (agg buffer is 25.6MB < 192MB L2).
__global__ __launch_bounds__(256, 1)
void gin_scatter_kernel(const float* __restrict__ x, const int* __restrict__ src,
                        const int* __restrict__ dst, float* __restrict__ agg,
                        int nedges) {
    const long gid = (long)blockIdx.x * 256 + threadIdx.x;
    const long e = gid >> 5;
    const int lane2 = (int)(gid & 31) * 2;
    if (e < (long)nedges) {
        const int s = src[e];
        const int d = dst[e];
        const float2 v = *(const float2*)(x + (long)s * 64 + lane2);
        atomicAdd(agg + (long)d * 64 + lane2 + 0, v.x);
        atomicAdd(agg + (long)d * 64 + lane2 + 1, v.y);
    }
}

// ---------------------------------------------------------------- fused MLP
// 128 threads = 4 wave32; each wave owns a 16x64 row tile.
// f32 WMMA 16x16x4 layouts (ISA 7.12.2):
//   A 16x4:  lane l -> M = l%16 ; lane-half selects K {0,1} vs {2,3}; 2 VGPRs = K pair
//   B 4x16:  mirror of A (lane-half selects K+0/K+2, VGPR selects K+0/K+1, col = l%16)
//   C 16x16: VGPR g -> M = g + 8*(l/16), N = l%16
//
// Weights are staged in LDS K-pair-interleaved:
//   sW[(k>>1)*WPSTR + n*2 + (k&1)] = W[k][n]
// so each B fragment is ONE aligned ds_load_b64 straight into a WMMA-ready
// register pair (no fixup moves).  WPSTR=132 (=128+4) de-conflicts the two
// lane halves (k-pair rows 2k vs 2k+1) across LDS banks.
constexpr int MLP_WAVES = 4;           // 128 threads, 64 rows / block
constexpr int TSTR  = 68;              // tile row stride (floats): bank-spread, 16B-aligned
constexpr int WPSTR = 132;             // weight k-pair row stride (floats)

__global__ __launch_bounds__(128, 1)
void gin_mlp_kernel(const float* __restrict__ hin,
                    const float* __restrict__ wa, const float* __restrict__ ba,
                    const float* __restrict__ wb, const float* __restrict__ bb,
                    float* __restrict__ out, int nrows) {
    __shared__ float sWA[32 * WPSTR];
    __shared__ float sWB[32 * WPSTR];
    __shared__ float sBA[64];
    __shared__ float sBB[64];
    __shared__ float sTile[MLP_WAVES][16 * TSTR];

    const int tid = threadIdx.x;

    // Stage weights K-pair interleaved + biases.
    for (int j = tid; j < 4096; j += 128) {
        const int k = j >> 6, n = j & 63;
        const int off = (k >> 1) * WPSTR + n * 2 + (k & 1);
        sWA[off] = wa[j];
        sWB[off] = wb[j];
    }
    if (tid < 64) { sBA[tid] = ba[tid]; sBB[tid] = bb[tid]; }

    const int wave = tid >> 5;
    const int lane = tid & 31;
    const int col  = lane & 15;
    const int half = lane >> 4;
    const long row0 = (long)blockIdx.x * (MLP_WAVES * 16) + wave * 16;

    float* tile = &sTile[wave][0];

    // Coalesced stage of this wave's 16x64 input tile (rows clamped for tail block).
    #pragma unroll
    for (int i = 0; i < 8; ++i) {
        const int idx = i * 128 + lane * 4;      // 0..1023 in float4 steps
        const int r = idx >> 6, c = idx & 63;
        long gr = row0 + r;
        if (gr > (long)nrows - 1) gr = (long)nrows - 1;
        const float4 v = *(const float4*)(hin + gr * 64 + c);
        *(float4*)(tile + r * TSTR + c) = v;
    }
    __syncthreads();   // weights visible to all waves

    // ---- GEMM1: tmp = relu(tile @ WA + BA) ----
    v2f afrag[16];
    #pragma unroll
    for (int k = 0; k < 16; ++k) {
        const int kk = k * 4 + half * 2;
        afrag[k] = *(const v2f*)(tile + col * TSTR + kk);
    }

    #pragma unroll
    for (int nt = 0; nt < 4; ++nt) {
        const float bias = sBA[nt * 16 + col];
        v8f acc;
        #pragma unroll
        for (int g = 0; g < 8; ++g) acc[g] = bias;
        #pragma unroll
        for (int k = 0; k < 16; ++k) {
            // B fragment: K-pair (k*4+half*2, +1), column nt*16+col — one b64 load.
            const v2f b = *(const v2f*)(sWA + (k * 2 + half) * WPSTR + (nt * 16 + col) * 2);
            acc = __builtin_amdgcn_wmma_f32_16x16x4_f32(
                false, afrag[k], false, b, (short)0, acc, false, false);
        }
        // ReLU, overwrite tile with tmp (A-frags already in regs; region is
        // wave-private so no block barrier needed, DS ops are in-order per wave).
        #pragma unroll
        for (int g = 0; g < 8; ++g) {
            const float v = acc[g] > 0.0f ? acc[g] : 0.0f;
            tile[(g + half * 8) * TSTR + nt * 16 + col] = v;
        }
    }

    // ---- GEMM2: out = tmp @ WB + BB ----
    #pragma unroll
    for (int k = 0; k < 16; ++k) {
        const int kk = k * 4 + half * 2;
        afrag[k] = *(const v2f*)(tile + col * TSTR + kk);
    }

    #pragma unroll
    for (int nt = 0; nt < 4; ++nt) {
        const float bias = sBB[nt * 16 + col];
        v8f acc;
        #pragma unroll
        for (int g = 0; g < 8; ++g) acc[g] = bias;
        #pragma unroll
        for (int k = 0; k < 16; ++k) {
            const v2f b = *(const v2f*)(sWB + (k * 2 + half) * WPSTR + (nt * 16 + col) * 2);
            acc = __builtin_amdgcn_wmma_f32_16x16x4_f32(
                false, afrag[k], false, b, (short)0, acc, false, false);
        }
        #pragma unroll
        for (int g = 0; g < 8; ++g) {
            const long r = row0 + g + half * 8;
            if (r < (long)nrows) out[r * 64 + nt * 16 + col] = acc[g];
        }
    }
}

// ---------------------------------------------------------------- launch
extern "C" void kernel_launch(void* const* d_in, const int* in_sizes, int n_in,
                              void* d_out, int out_size, void* d_ws, size_t ws_size,
                              hipStream_t stream) {
    const float* x    = (const float*)d_in[0];
    const int*   ei   = (const int*)d_in[1];
    const float* eps1 = (const float*)d_in[2];
    const float* w1a  = (const float*)d_in[3];
    const float* b1a  = (const float*)d_in[4];
    const float* w1b  = (const float*)d_in[5];
    const float* b1b  = (const float*)d_in[6];
    const float* eps2 = (const float*)d_in[7];
    const float* w2a  = (const float*)d_in[8];
    const float* b2a  = (const float*)d_in[9];
    const float* w2b  = (const float*)d_in[10];
    const float* b2b  = (const float*)d_in[11];
    float* out = (float*)d_out;

    const int nodes = in_sizes[0] / 64;   // N = 100000
    const int edges = in_sizes[1] / 2;    // E = 1600000
    const int* src = ei;
    const int* dst = ei + edges;

    float* s1 = (float*)d_ws;                        // h_pre buffer  [N,64]
    float* s2 = s1 + (size_t)nodes * 64;             // hidden buffer [N,64]

    const int nvec4 = nodes * 16;                    // N*64/4 float4s
    const int scale_grid   = (nvec4 + 255) / 256;
    const long scat_threads = (long)edges * 32;
    const int scatter_grid = (int)((scat_threads + 255) / 256);
    const int mlp_grid     = (nodes + MLP_WAVES * 16 - 1) / (MLP_WAVES * 16);

    // ---- layer 1 ----
    gin_scale_kernel<<<scale_grid, 256, 0, stream>>>(x, eps1, s1, nvec4);
    gin_scatter_kernel<<<scatter_grid, 256, 0, stream>>>(x, src, dst, s1, edges);
    gin_mlp_kernel<<<mlp_grid, 128, 0, stream>>>(s1, w1a, b1a, w1b, b1b, s2, nodes);

    // ---- layer 2 ----
    gin_scale_kernel<<<scale_grid, 256, 0, stream>>>(s2, eps2, s1, nvec4);
    gin_scatter_kernel<<<scatter_grid, 256, 0, stream>>>(s2, src, dst, s1, edges);
    gin_mlp_kernel<<<mlp_grid, 128, 0, stream>>>(s1, w2a, b2a, w2b, b2b, out, nodes);
}